// CannyDetector_9277129359558
// MI455X (gfx1250) — compile-verified
//
#include <hip/hip_runtime.h>
#include <cstdint>

// ---------------------------------------------------------------------------
// Canny on MI455X (gfx1250, wave32).
// Bandwidth-bound: fuse blur+sobel+mag+orient so the 96MB blurred image never
// hits DRAM; vertical Gaussian runs as banded-matrix WMMA (f32 16x16x4);
// interior image tiles are fetched by the Tensor Data Mover (TDM) into LDS.
// Workspace: mag 32MB f32 | dir 8MB u8 | class 8MB u8  (48 MB total).
// ---------------------------------------------------------------------------

#define HH 1024
#define WW 1024
#define NB 8

#define INNER_ROWS 14   // mag rows produced per workgroup (16 - 2 halo)
#define INNER_COLS 126  // mag cols produced per workgroup (128 - 2 halo)

typedef float v2f  __attribute__((ext_vector_type(2)));
typedef float v8f  __attribute__((ext_vector_type(8)));
typedef unsigned int u32x4 __attribute__((ext_vector_type(4)));
typedef int   i32x4 __attribute__((ext_vector_type(4)));
typedef int   i32x8 __attribute__((ext_vector_type(8)));

// scipy.signal.gaussian(5, std=1), unnormalized: exp(-0.5*n^2), n=-2..2
#define G0 0.13533528323661270f
#define G1 0.60653065971263342f
#define G2 1.0f

__device__ __forceinline__ float g5(int d) {
    // banded-matrix entry: G[d] for d in [0,5), else 0 (select chain, no scratch)
    float r = 0.0f;
    r = (d == 0) ? G0 : r;
    r = (d == 1) ? G1 : r;
    r = (d == 2) ? G2 : r;
    r = (d == 3) ? G1 : r;
    r = (d == 4) ? G0 : r;
    return r;
}

// ---------------------------------------------------------------------------
// Kernel 1: fused separable Gaussian blur (vertical pass on the WMMA pipe) +
// Sobel + gradient magnitude (sum over 3 channels) + quantized orientation.
// Workgroup = 256 threads = 8 waves; blurred tile 16 rows x 128 cols in LDS,
// one 16x16 WMMA tile per wave.  Out-of-image blurred values are forced to 0
// (zero-pad semantics of the reference's sobel conv).
// ---------------------------------------------------------------------------
__global__ __launch_bounds__(256)
void canny_blur_sobel_kernel(const float* __restrict__ img,
                             float* __restrict__ mag,
                             uint8_t* __restrict__ dir8)
{
    __shared__ float sImg[20][132];        // raw image tile (+2 halo each side)
    __shared__ float sH[20][128];          // after horizontal 5-tap
    __shared__ float sBlur[3][16][128];    // blurred, all 3 channels

    const int tid  = threadIdx.x;
    const int lane = tid & 31;
    const int wv   = tid >> 5;        // wave id 0..7 -> 16-col sub-tile
    const int hi   = lane >> 4;       // half-wave (0/1)
    const int ln   = lane & 15;

    const int b  = blockIdx.z;
    const int r0 = (int)blockIdx.y * INNER_ROWS - 1;   // blurred-tile top row
    const int c0 = (int)blockIdx.x * INNER_COLS - 1;   // blurred-tile left col

    // TDM handles right/bottom overrun (reads past tensor dims return 0);
    // only tiles whose halo goes negative need the generic zero-pad path.
    const bool tdm_ok = (blockIdx.x > 0) && (blockIdx.y > 0);

    for (int ch = 0; ch < 3; ++ch) {
        __syncthreads();   // protect sImg/sH reuse across channel iterations
        const float* plane = img + (size_t)(b * 3 + ch) * (HH * WW);

        // -- stage 1: fetch 20x132 image tile into LDS ----------------------
        if (tdm_ok) {
            // Tensor Data Mover: 2-D tile (20 rows x 132 f32) from the
            // 1024x1024 plane straight into LDS; one issue from wave 0,
            // tracked by TENSORcnt, published to the workgroup by the barrier.
            if (tid < 32) {
                const float* tile0 = plane + (size_t)(r0 - 2) * WW + (c0 - 2);
                uint64_t ga = (uint64_t)(uintptr_t)tile0;            // 57-bit VA
                uint32_t lds = (uint32_t)(uintptr_t)&sImg[0][0];     // LDS byte offset

                u32x4 g0;
                g0[0] = 1u;                                  // count=1, user mode
                g0[1] = lds;                                 // lds_addr
                g0[2] = (uint32_t)ga;                        // global_addr[31:0]
                g0[3] = (uint32_t)((ga >> 32) & 0x01FFFFFFu) // global_addr[56:32]
                      | 0x80000000u;                         // type=2 ("image")

                i32x8 g1;
                g1[0] = 0x00020000;          // workgroup_mask=0, data_size=2 (4B)
                g1[1] = (int)(WW << 16);     // tensor_dim0[15:0] in bits 63:48
                g1[2] = (int)(HH << 16);     // dim0 hi=0 | tensor_dim1[15:0]
                g1[3] = (int)(132u << 16);   // dim1 hi=0 | tile_dim0=132
                g1[4] = 20;                  // tile_dim1=20, tile_dim2=0
                g1[5] = WW;                  // tensor_dim0_stride = 1024
                g1[6] = 0;                   // stride hi | dim1_stride[15:0]
                g1[7] = (HH * WW) >> 16;     // tensor_dim1_stride hi (unused: 2-D)

                i32x4 gz  = {0, 0, 0, 0};            // groups 2/3: tile_dim3/4 = 0
                i32x8 gz8 = {0, 0, 0, 0, 0, 0, 0, 0};
                __builtin_amdgcn_tensor_load_to_lds(g0, g1, gz, gz, gz8, 0);
                __builtin_amdgcn_s_wait_tensorcnt(0);
            }
        } else {
            // border tiles: generic zero-padded load
            for (int i = tid; i < 20 * 132; i += 256) {
                int rr = i / 132, cc = i - rr * 132;
                int gr = r0 - 2 + rr, gc = c0 - 2 + cc;
                float v = 0.0f;
                if ((unsigned)gr < HH && (unsigned)gc < WW) v = plane[(size_t)gr * WW + gc];
                sImg[rr][cc] = v;
            }
        }
        __syncthreads();

        // -- stage 2: horizontal 5-tap (trivial VALU work) ------------------
        for (int i = tid; i < 20 * 128; i += 256) {
            int rr = i >> 7, cc = i & 127;
            sH[rr][cc] = G0 * sImg[rr][cc]     + G1 * sImg[rr][cc + 1]
                       + G2 * sImg[rr][cc + 2] + G1 * sImg[rr][cc + 3]
                       + G0 * sImg[rr][cc + 4];
        }
        __syncthreads();

        // -- stage 3: vertical 5-tap as Vband(16x20) @ H(20x16) on WMMA -----
        // A layout (16x4 f32): lane m=ln, VGPR0 -> K=2*hi, VGPR1 -> K=2*hi+1
        // B layout (4x16 f32): VGPR v, half-wave h -> row K = v + 2h, col N=ln
        v8f acc = {};
        #pragma unroll
        for (int j = 0; j < 5; ++j) {
            const int ka = 4 * j + 2 * hi;
            v2f a, bm;
            a.x = g5(ka - ln);          // Vband[m][ka]   = G[ka - m]
            a.y = g5(ka + 1 - ln);      // Vband[m][ka+1]
            bm.x = sH[ka][16 * wv + ln];
            bm.y = sH[ka + 1][16 * wv + ln];
            acc = __builtin_amdgcn_wmma_f32_16x16x4_f32(
                false, a, false, bm, (short)0, acc, false, false);
        }

        // -- stage 4: D layout: component v of lane -> (M = v + 8*hi, N = ln)
        #pragma unroll
        for (int v = 0; v < 8; ++v) {
            int m2 = v + 8 * hi;
            int gr = r0 + m2, gc = c0 + 16 * wv + ln;
            float val = ((unsigned)gr < HH && (unsigned)gc < WW) ? acc[v] : 0.0f;
            sBlur[ch][m2][16 * wv + ln] = val;   // 0 outside image = conv zero-pad
        }
    }
    __syncthreads();

    // -- Sobel + magnitude + orientation on the inner 14x126 region ---------
    for (int i = tid; i < INNER_ROWS * INNER_COLS; i += 256) {
        int rr = 1 + i / INNER_COLS, cc = 1 + i % INNER_COLS;
        int gr = r0 + rr, gc = c0 + cc;              // gr,gc >= 0 by construction
        if (gr >= HH || gc >= WW) continue;
        float magv = 0.0f, sgx = 0.0f, sgy = 0.0f;
        #pragma unroll
        for (int ch = 0; ch < 3; ++ch) {
            float a00 = sBlur[ch][rr - 1][cc - 1], a01 = sBlur[ch][rr - 1][cc], a02 = sBlur[ch][rr - 1][cc + 1];
            float a10 = sBlur[ch][rr][cc - 1],                                  a12 = sBlur[ch][rr][cc + 1];
            float a20 = sBlur[ch][rr + 1][cc - 1], a21 = sBlur[ch][rr + 1][cc], a22 = sBlur[ch][rr + 1][cc + 1];
            float gx = (a00 - a02) + 2.0f * (a10 - a12) + (a20 - a22);
            float gy = (a00 + 2.0f * a01 + a02) - (a20 + 2.0f * a21 + a22);
            magv += sqrtf(gx * gx + gy * gy);
            sgx += gx; sgy += gy;
        }
        float orient = atan2f(sgy, sgx) * 57.29577951308232f + 180.0f;
        int k = (int)rintf(orient * (1.0f / 45.0f));     // round-half-even, like jnp.round
        size_t o = (size_t)b * (HH * WW) + (size_t)gr * WW + gc;
        mag[o]  = magv;
        dir8[o] = (uint8_t)(k & 7);
    }
}

// ---------------------------------------------------------------------------
// Kernel 2: NMS + classify.  Faithful to the reference's flat gather:
// with B==8, flat[idx*pc + pix] == all_filtered[batch=idx, channel=b, pixel],
// i.e. directional filter #b (the BATCH index) applied to grad_mag of image
// `idx` — a cross-image gather.  grad_mag (32MB) is L2-resident on MI455X.
// class: 0 = below low, 1 = [low, high], 2 = above high.
// ---------------------------------------------------------------------------
__global__ __launch_bounds__(256)
void canny_nms_kernel(const float* __restrict__ mag,
                      const uint8_t* __restrict__ dir8,
                      uint8_t* __restrict__ cls)
{
    const int PC = HH * WW;
    size_t gid = (size_t)blockIdx.x * 256 + threadIdx.x;
    if (gid >= (size_t)NB * PC) return;
    int b = (int)(gid >> 20);            // PC = 2^20
    int p = (int)(gid & (PC - 1));
    int y = p >> 10, x = p & (WW - 1);

    // dir_f[b]: +1 at center, -1 at (dy,dx); packed 2-bit LUTs (value+1)
    int dx = (int)((0x901Au >> (2 * b)) & 3u) - 1;   // {1,1,0,-1,-1,-1,0,1}
    int dy = (int)((0x01A9u >> (2 * b)) & 3u) - 1;   // {0,1,1,1,0,-1,-1,-1}

    int ip = dir8[gid];
    int in_ = (ip + 4) & 7;
    int yn = y + dy, xn = x + dx;
    bool nb_ok = ((unsigned)yn < HH) && ((unsigned)xn < WW);  // conv zero-pad
    int pn = yn * WW + xn;

    float sp = mag[(size_t)ip * PC + p]  - (nb_ok ? mag[(size_t)ip * PC + pn]  : 0.0f);
    float sn = mag[(size_t)in_ * PC + p] - (nb_ok ? mag[(size_t)in_ * PC + pn] : 0.0f);
    float thin = (fminf(sp, sn) > 0.0f) ? mag[gid] : 0.0f;

    uint8_t c = 0;
    if (thin > 5.0f)       c = 2;   // HIGH_T
    else if (thin >= 2.5f) c = 1;   // LOW_T
    cls[gid] = c;
}

// ---------------------------------------------------------------------------
// Kernel 3: hysteresis-lite + border zero.  The reference's threshold/connect
// algebra reduces to: out = border ? 0 : strong ? 1
//                         : (mid && any 8-neighbor strong) ? 1 : 0
// ---------------------------------------------------------------------------
__global__ __launch_bounds__(256)
void canny_final_kernel(const uint8_t* __restrict__ cls, float* __restrict__ out)
{
    const int PC = HH * WW;
    size_t gid = (size_t)blockIdx.x * 256 + threadIdx.x;
    if (gid >= (size_t)NB * PC) return;
    int p = (int)(gid & (PC - 1));
    int y = p >> 10, x = p & (WW - 1);

    float r = 0.0f;
    if (y > 0 && y < HH - 1 && x > 0 && x < WW - 1) {
        uint8_t c = cls[gid];
        if (c == 2) {
            r = 1.0f;
        } else if (c == 1) {
            bool any =
                cls[gid - WW - 1] == 2 || cls[gid - WW] == 2 || cls[gid - WW + 1] == 2 ||
                cls[gid - 1]      == 2 ||                       cls[gid + 1]      == 2 ||
                cls[gid + WW - 1] == 2 || cls[gid + WW] == 2 || cls[gid + WW + 1] == 2;
            if (any) r = 1.0f;
        }
    }
    out[gid] = r;
}

// ---------------------------------------------------------------------------
extern "C" void kernel_launch(void* const* d_in, const int* in_sizes, int n_in,
                              void* d_out, int out_size, void* d_ws, size_t ws_size,
                              hipStream_t stream)
{
    (void)in_sizes; (void)n_in; (void)out_size; (void)ws_size;
    const float* img = (const float*)d_in[0];     // (8,3,1024,1024) f32

    const size_t npix = (size_t)NB * HH * WW;     // 8M pixels
    float*   mag  = (float*)d_ws;                                   // 32 MB
    uint8_t* dir8 = (uint8_t*)d_ws + npix * sizeof(float);          //  8 MB
    uint8_t* cls  = dir8 + npix;                                    //  8 MB
    float*   out  = (float*)d_out;

    dim3 grid1((WW + INNER_COLS - 1) / INNER_COLS,    // 9
               (HH + INNER_ROWS - 1) / INNER_ROWS,    // 74
               NB);                                   // 8
    canny_blur_sobel_kernel<<<grid1, 256, 0, stream>>>(img, mag, dir8);

    int nblk = (int)((npix + 255) / 256);
    canny_nms_kernel<<<nblk, 256, 0, stream>>>(mag, dir8, cls);
    canny_final_kernel<<<nblk, 256, 0, stream>>>(cls, out);
}